// AttentionBlock_53257594470526
// MI455X (gfx1250) — compile-verified
//
#include <hip/hip_runtime.h>
#include <math.h>

#define F_IN   128
#define HC     128
#define HEADS  8
#define C_OUT  16
#define NEG_SLOPE 0.2f
#define LN_EPS    1e-5f

typedef __attribute__((ext_vector_type(2))) float v2f;
typedef __attribute__((ext_vector_type(8))) float v8f;
typedef __attribute__((ext_vector_type(4))) int   v4i;

#if defined(__has_builtin)
#if __has_builtin(__builtin_amdgcn_global_load_async_to_lds_b128)
#define HAVE_ASYNC_LDS 1
#endif
#endif

#ifdef HAVE_ASYNC_LDS
// builtin signature: (v4i AS1* gaddr, v4i AS3* ldsaddr, imm offset, imm cpol)
typedef __attribute__((address_space(1))) v4i* gv4_ptr;
typedef __attribute__((address_space(3))) v4i* lv4_ptr;
#endif

// ---------- helpers ----------
__device__ __forceinline__ float atomic_max_float(float* addr, float val) {
  // IEEE-754 order-preserving integer punning (valid incl. -inf init)
  if (val >= 0.0f)
    return __int_as_float(atomicMax((int*)addr, __float_as_int(val)));
  else
    return __uint_as_float(atomicMin((unsigned int*)addr, __float_as_uint(val)));
}

__device__ __forceinline__ float leaky(float t) {
  return (t > 0.0f) ? t : NEG_SLOPE * t;
}

// ---------- fp32 WMMA GEMM: D[M,128] = A[M,128] @ B[128,128] (+bias)(+residual) ----------
// Block = 256 threads = 8 waves. Block tile: 64 rows x 128 cols.
// Wave w owns columns [16w,16w+16) and FOUR 16-row M-subtiles (4 accumulators),
// so each B operand pair feeds 4 WMMAs (4x operand reuse vs 1 tile/wave).
// A tile staged in LDS (row stride 132 floats -> conflict-free column reads, 16B-aligned rows).
__global__ __launch_bounds__(256) void gemm128_wmma(
    const float* __restrict__ A, const float* __restrict__ Bm,
    float* __restrict__ D, int M,
    const float* __restrict__ bias, const float* __restrict__ residual) {
  __shared__ __align__(16) float As[64][132];
  const int m0  = blockIdx.x * 64;
  const int tid = threadIdx.x;

  const bool full_tile = (m0 + 64 <= M);
#ifdef HAVE_ASYNC_LDS
  if (full_tile) {
    // Direct global->LDS async copy (GLOBAL_LOAD_ASYNC_TO_LDS_B128, ASYNCcnt).
    // LDS operand: flat LDS address truncates to the 32-bit LDS offset (ISA 10.2).
#pragma unroll
    for (int i = 0; i < 8; ++i) {
      int e4 = tid + i * 256;            // float4 index, 0..2047
      int r  = e4 >> 5;                  // 32 float4 per 128-col row
      int c  = (e4 & 31) * 4;
      gv4_ptr g = (gv4_ptr)(uintptr_t)(A + (size_t)(m0 + r) * 128 + c);
      lv4_ptr l = (lv4_ptr)(uint32_t)(uintptr_t)&As[r][c];
      __builtin_amdgcn_global_load_async_to_lds_b128(g, l, 0, 0);
    }
    asm volatile("s_wait_asynccnt 0" ::: "memory");
  } else
#endif
  {
#pragma unroll
    for (int i = 0; i < 8; ++i) {
      int e4  = tid + i * 256;
      int r   = e4 >> 5;
      int c   = (e4 & 31) * 4;
      int row = m0 + r;
      float4 v = make_float4(0.f, 0.f, 0.f, 0.f);
      if (row < M) v = *(const float4*)(A + (size_t)row * 128 + c);
      *(float4*)&As[r][c] = v;
    }
  }
  __syncthreads();

  const int wave = tid >> 5;
  const int lane = tid & 31;
  const int n0   = wave * 16;
  const int l15  = lane & 15;
  const int koff = (lane >> 4) << 1;   // lanes 0-15 -> K=0,1 ; lanes 16-31 -> K=2,3

  v8f acc[4] = {};
#pragma unroll
  for (int k = 0; k < 128; k += 4) {
    v2f b;
    b[0] = Bm[(k + koff)     * 128 + n0 + l15];
    b[1] = Bm[(k + koff + 1) * 128 + n0 + l15];
#pragma unroll
    for (int s = 0; s < 4; ++s) {
      v2f a;
      a[0] = As[s * 16 + l15][k + koff];
      a[1] = As[s * 16 + l15][k + koff + 1];
      acc[s] = __builtin_amdgcn_wmma_f32_16x16x4_f32(
          /*neg_a=*/false, a, /*neg_b=*/false, b,
          /*c_mod=*/(short)0, acc[s], /*reuse_a=*/false, /*reuse_b=*/false);
    }
  }

  // C/D layout: VGPR g holds M = g (lanes 0-15) / g+8 (lanes 16-31), N = lane&15
  const int rbase = (lane >> 4) << 3;
  const int col   = n0 + l15;
#pragma unroll
  for (int s = 0; s < 4; ++s) {
#pragma unroll
    for (int g = 0; g < 8; ++g) {
      int row = m0 + s * 16 + rbase + g;
      if (row < M) {
        float v = acc[s][g];
        if (bias)     v += bias[col];
        if (residual) v += residual[(size_t)row * 128 + col];
        D[(size_t)row * 128 + col] = v;
      }
    }
  }
}

// ---------- workspace init ----------
__global__ void init_ws(float* __restrict__ agg, float* __restrict__ emax,
                        float* __restrict__ denom, int N) {
  int tid = blockIdx.x * blockDim.x + threadIdx.x;
  if (tid < N * HC) agg[tid] = 0.0f;
  if (tid < N * HEADS) {
    emax[tid]  = __int_as_float(0xff800000);  // -inf
    denom[tid] = 0.0f;
  }
}

// ---------- per-(node,head) attention logits: a = <h[n,head,:], att[head,:]> ----------
__global__ void att_logits(const float* __restrict__ h,
                           const float* __restrict__ att_src,
                           const float* __restrict__ att_dst,
                           float* __restrict__ a_s, float* __restrict__ a_d, int N) {
  int tid = blockIdx.x * blockDim.x + threadIdx.x;
  if (tid >= N * HEADS) return;
  int n = tid >> 3, head = tid & 7;
  const float* hp = h + (size_t)n * HC + head * C_OUT;
  const float* sp = att_src + head * C_OUT;
  const float* dp = att_dst + head * C_OUT;
  float s = 0.0f, d = 0.0f;
#pragma unroll
  for (int c = 0; c < C_OUT; ++c) {
    float v = hp[c];
    s += v * sp[c];
    d += v * dp[c];
  }
  a_s[tid] = s;
  a_d[tid] = d;
}

// ---------- segment softmax pass 1: per-dst max of leaky(logit) ----------
__global__ void edge_max_kernel(const int* __restrict__ src, const int* __restrict__ dst,
                                int E, int N,
                                const float* __restrict__ a_s, const float* __restrict__ a_d,
                                float* __restrict__ emax) {
  int tid = blockIdx.x * blockDim.x + threadIdx.x;
  if (tid >= E + N) return;
  int s, d;
  if (tid < E) { s = src[tid]; d = dst[tid]; } else { s = tid - E; d = s; }
#pragma unroll
  for (int head = 0; head < HEADS; ++head) {
    float t = leaky(a_s[s * HEADS + head] + a_d[d * HEADS + head]);
    atomic_max_float(&emax[d * HEADS + head], t);
  }
}

// ---------- segment softmax pass 2: per-dst sum of exp ----------
__global__ void edge_sum_kernel(const int* __restrict__ src, const int* __restrict__ dst,
                                int E, int N,
                                const float* __restrict__ a_s, const float* __restrict__ a_d,
                                const float* __restrict__ emax, float* __restrict__ denom) {
  int tid = blockIdx.x * blockDim.x + threadIdx.x;
  if (tid >= E + N) return;
  int s, d;
  if (tid < E) { s = src[tid]; d = dst[tid]; } else { s = tid - E; d = s; }
#pragma unroll
  for (int head = 0; head < HEADS; ++head) {
    float t  = leaky(a_s[s * HEADS + head] + a_d[d * HEADS + head]);
    float ex = __expf(t - emax[d * HEADS + head]);
    atomicAdd(&denom[d * HEADS + head], ex);
  }
}

// ---------- pass 3: weighted scatter-aggregate, one wave32 per edge ----------
// Lane l covers channels [4l, 4l+4) of HC=128 -> head = l>>2; atomics resolve in L2
// (agg = 25.6 MB << 192 MB global L2).
__global__ __launch_bounds__(256) void edge_aggregate(
    const int* __restrict__ src, const int* __restrict__ dst, int E, int N,
    const float* __restrict__ a_s, const float* __restrict__ a_d,
    const float* __restrict__ emax, const float* __restrict__ denom,
    const float* __restrict__ h, float* __restrict__ agg) {
  int gid  = blockIdx.x * blockDim.x + threadIdx.x;
  int eidx = gid >> 5;
  int lane = gid & 31;
  if (eidx >= E + N) return;
  int s, d;
  if (eidx < E) { s = src[eidx]; d = dst[eidx]; } else { s = eidx - E; d = s; }

  int head    = lane >> 2;
  float t     = leaky(a_s[s * HEADS + head] + a_d[d * HEADS + head]);
  float alpha = __expf(t - emax[d * HEADS + head]) / denom[d * HEADS + head];

  const float4 hv = *(const float4*)(h + (size_t)s * HC + lane * 4);
  float* ap = agg + (size_t)d * HC + lane * 4;
  atomicAdd(ap + 0, alpha * hv.x);
  atomicAdd(ap + 1, alpha * hv.y);
  atomicAdd(ap + 2, alpha * hv.z);
  atomicAdd(ap + 3, alpha * hv.w);
}

// ---------- +conv_bias then LayerNorm (in place), one wave32 per node ----------
__global__ __launch_bounds__(256) void layernorm_bias(
    float* __restrict__ agg, const float* __restrict__ conv_bias,
    const float* __restrict__ ln_g, const float* __restrict__ ln_b, int N) {
  int gid  = blockIdx.x * blockDim.x + threadIdx.x;
  int n    = gid >> 5;
  int lane = gid & 31;
  if (n >= N) return;

  float4 v = *(float4*)(agg + (size_t)n * HC + lane * 4);
  const float4 cb = *(const float4*)(conv_bias + lane * 4);
  v.x += cb.x; v.y += cb.y; v.z += cb.z; v.w += cb.w;

  float sum = v.x + v.y + v.z + v.w;
  float sq  = v.x * v.x + v.y * v.y + v.z * v.z + v.w * v.w;
#pragma unroll
  for (int off = 16; off; off >>= 1) {
    sum += __shfl_xor(sum, off, 32);
    sq  += __shfl_xor(sq,  off, 32);
  }
  float mu  = sum * (1.0f / 128.0f);
  float var = sq  * (1.0f / 128.0f) - mu * mu;
  float r   = rsqrtf(var + LN_EPS);

  const float4 g = *(const float4*)(ln_g + lane * 4);
  const float4 b = *(const float4*)(ln_b + lane * 4);
  v.x = (v.x - mu) * r * g.x + b.x;
  v.y = (v.y - mu) * r * g.y + b.y;
  v.z = (v.z - mu) * r * g.z + b.z;
  v.w = (v.w - mu) * r * g.w + b.w;
  *(float4*)(agg + (size_t)n * HC + lane * 4) = v;
}

// ---------- launch ----------
extern "C" void kernel_launch(void* const* d_in, const int* in_sizes, int n_in,
                              void* d_out, int out_size, void* d_ws, size_t ws_size,
                              hipStream_t stream) {
  const float* x         = (const float*)d_in[0];
  const int*   edge_idx  = (const int*)d_in[1];
  const float* W         = (const float*)d_in[2];
  const float* att_src   = (const float*)d_in[3];
  const float* att_dst   = (const float*)d_in[4];
  const float* conv_bias = (const float*)d_in[5];
  const float* ln_g      = (const float*)d_in[6];
  const float* ln_b      = (const float*)d_in[7];
  const float* ff_w      = (const float*)d_in[8];
  const float* ff_b      = (const float*)d_in[9];
  float*       out       = (float*)d_out;

  const int N = in_sizes[0] / F_IN;
  const int E = in_sizes[1] / 2;
  const int* src = edge_idx;
  const int* dst = edge_idx + E;

  // workspace layout (floats): h[N*128] | agg[N*128] | a_s[N*8] | a_d[N*8] | emax[N*8] | denom[N*8]
  float* ws    = (float*)d_ws;
  float* h     = ws;
  float* agg   = h   + (size_t)N * HC;
  float* a_s   = agg + (size_t)N * HC;
  float* a_d   = a_s + (size_t)N * HEADS;
  float* emax  = a_d + (size_t)N * HEADS;
  float* denom = emax + (size_t)N * HEADS;

  const int mtiles = (N + 63) / 64;
  const int total  = E + N;

  init_ws<<<(N * HC + 255) / 256, 256, 0, stream>>>(agg, emax, denom, N);
  gemm128_wmma<<<mtiles, 256, 0, stream>>>(x, W, h, N, nullptr, nullptr);
  att_logits<<<(N * HEADS + 255) / 256, 256, 0, stream>>>(h, att_src, att_dst, a_s, a_d, N);
  edge_max_kernel<<<(total + 255) / 256, 256, 0, stream>>>(src, dst, E, N, a_s, a_d, emax);
  edge_sum_kernel<<<(total + 255) / 256, 256, 0, stream>>>(src, dst, E, N, a_s, a_d, emax, denom);
  edge_aggregate<<<((size_t)total * 32 + 255) / 256, 256, 0, stream>>>(
      src, dst, E, N, a_s, a_d, emax, denom, h, agg);
  layernorm_bias<<<((size_t)N * 32 + 255) / 256, 256, 0, stream>>>(agg, conv_bias, ln_g, ln_b, N);
  gemm128_wmma<<<mtiles, 256, 0, stream>>>(agg, ff_w, out, N, ff_b, x);
}